// LinearAttention_82652350644602
// MI455X (gfx1250) — compile-verified
//
#include <hip/hip_runtime.h>

// CDNA5 / gfx1250: wave32, WMMA f32 16x16x4 path.
typedef __attribute__((ext_vector_type(2))) float v2f;
typedef __attribute__((ext_vector_type(8))) float v8f;

#define NN 65536
#define DD 1024
#define CHUNKS 64                    // n-chunks for phase-1 partials
#define ROWS_PER_CHUNK (NN / CHUNKS) // 1024 rows per chunk
#define DTILES (DD / 16)             // 64 d-tiles of width 16

// ---------------------------------------------------------------------------
// Phase 1: partial[chunk][d] = sum over chunk rows of g[n] * P[n][d]
// One wave per (chunk, d-tile). Per iteration, one V_WMMA_F32_16X16X4_F32:
//   A (16x4): gradient values g[n..n+3] broadcast into every M row
//   B (4x16): past_states rows n..n+3, columns d0..d0+15
// => every row of D accumulates the same partial dot products; we keep row 0.
// A layout (32-bit 16x4): lanes 0-15 -> {K0,K1}, lanes 16-31 -> {K2,K3}.
// B layout assumed analogous: vgpr0 = rows {K0 | K2}, vgpr1 = rows {K1 | K3}.
// ---------------------------------------------------------------------------
__global__ void __launch_bounds__(256)
la_phase1_gemv(const float* __restrict__ P, const float* __restrict__ g,
               float* __restrict__ partial) {
  const int wave  = (blockIdx.x * blockDim.x + threadIdx.x) >> 5;
  const int lane  = threadIdx.x & 31;
  const int chunk = wave >> 6;   // 0..63
  const int dtile = wave & 63;   // 0..63
  const int d0    = dtile * 16;
  const int n0    = chunk * ROWS_PER_CHUNK;
  const int half  = lane >> 4;   // 0: K rows 0/1, 1: K rows 2/3
  const int l16   = lane & 15;

  const float* pA = g + n0 + 2 * half;                          // g[n], g[n+1] (or +2/+3)
  const float* pB = P + (size_t)(n0 + 2 * half) * DD + d0 + l16; // P[n(+2)][d0+l16]

  v8f acc = {};
  for (int i = 0; i < ROWS_PER_CHUNK / 4; ++i) {
    v2f a; a.x = pA[0];  a.y = pA[1];   // K and K+1 gradient values
    v2f b; b.x = pB[0];  b.y = pB[DD];  // rows K and K+1, column d0+l16
    acc = __builtin_amdgcn_wmma_f32_16x16x4_f32(
        /*neg_a=*/false, a, /*neg_b=*/false, b,
        /*c_mod=*/(short)0, acc, /*reuse_a=*/false, /*reuse_b=*/false);
    pA += 4;
    pB += 4 * DD;
  }
  // C/D layout: VGPR0, lanes 0-15 hold (M=0, N=lane) -> the partial slice.
  if (lane < 16) partial[chunk * DD + d0 + l16] = acc[0];
}

// ---------------------------------------------------------------------------
// Phase 2: weighted[d] = sum over 64 chunks of partial[c][d]  (256 KB read)
// Fixed summation order -> deterministic across graph replays.
// ---------------------------------------------------------------------------
__global__ void __launch_bounds__(256)
la_phase2_reduce(const float* __restrict__ partial, float* __restrict__ weighted) {
  const int d = blockIdx.x * blockDim.x + threadIdx.x; // 0..1023
  float s = 0.f;
#pragma unroll 8
  for (int c = 0; c < CHUNKS; ++c) s += partial[c * DD + d];
  weighted[d] = s;
}

// ---------------------------------------------------------------------------
// Phase 3: out[m][n] = weighted[m] * cs[n], one 16x16 tile per wave via WMMA.
// A: column K=0 = weighted[m0..m0+15] (lanes 0-15), rest zero.
// B: row    K=0 = cs[n0..n0+15]       (lanes 0-15), rest zero.
// D[m][n] = weighted[m]*cs[n] exactly (C = 0).
// ---------------------------------------------------------------------------
__global__ void __launch_bounds__(256)
la_phase3_outer(const float* __restrict__ weighted, const float* __restrict__ cs,
                float* __restrict__ out) {
  const int wave = (blockIdx.x * blockDim.x + threadIdx.x) >> 5;
  const int lane = threadIdx.x & 31;
  const int tm   = wave >> 6;  // 0..63 row tile
  const int tn   = wave & 63;  // 0..63 col tile
  const int half = lane >> 4;
  const int l16  = lane & 15;

  v2f a; a.x = (half == 0) ? weighted[tm * 16 + l16] : 0.f; a.y = 0.f;
  v2f b; b.x = (half == 0) ? cs[tn * 16 + l16]       : 0.f; b.y = 0.f;
  v8f c = {};
  c = __builtin_amdgcn_wmma_f32_16x16x4_f32(
      false, a, false, b, (short)0, c, false, false);

  // D layout: VGPR j -> row (j + 8*half), col l16.
  float* o = out + (size_t)(tm * 16 + half * 8) * DD + tn * 16 + l16;
#pragma unroll
  for (int j = 0; j < 8; ++j) o[(size_t)j * DD] = c[j];
}

extern "C" void kernel_launch(void* const* d_in, const int* in_sizes, int n_in,
                              void* d_out, int out_size, void* d_ws, size_t ws_size,
                              hipStream_t stream) {
  (void)in_sizes; (void)n_in; (void)out_size; (void)ws_size;

  const float* past_states   = (const float*)d_in[0]; // [N, D]
  const float* current_state = (const float*)d_in[1]; // [D]
  const float* gradients     = (const float*)d_in[2]; // [N]
  float* out = (float*)d_out;                         // [D, D]

  float* partial  = (float*)d_ws;          // CHUNKS * DD floats (256 KB)
  float* weighted = partial + CHUNKS * DD; // DD floats (4 KB)

  // 4096 waves = 64 chunks x 64 d-tiles; 8 waves/block -> 512 blocks.
  la_phase1_gemv<<<(CHUNKS * DTILES) / 8, 256, 0, stream>>>(past_states, gradients, partial);
  la_phase2_reduce<<<DD / 256, 256, 0, stream>>>(partial, weighted);
  // 4096 waves = 64 x 64 output tiles.
  la_phase3_outer<<<(DTILES * DTILES) / 8, 256, 0, stream>>>(weighted, current_state, out);
}